// GCN_74706661146648
// MI455X (gfx1250) — compile-verified
//
#include <hip/hip_runtime.h>
#include <hip/hip_bf16.h>
#include <math.h>

typedef float v2f __attribute__((ext_vector_type(2)));
typedef float v8f __attribute__((ext_vector_type(8)));

// ---------------- degree / normalization ----------------

__global__ void k_init_deg(float* deg, int n) {
    int i = blockIdx.x * blockDim.x + threadIdx.x;
    if (i < n) deg[i] = 1.0f;  // self-loop
}

__global__ void k_deg_scatter(const int* __restrict__ dst, float* deg, int e) {
    int i = blockIdx.x * blockDim.x + threadIdx.x;
    if (i < e) atomicAdd(&deg[dst[i]], 1.0f);
}

__global__ void k_dinv(float* deg, int n) {
    int i = blockIdx.x * blockDim.x + threadIdx.x;
    if (i < n) deg[i] = rsqrtf(deg[i]);  // deg >= 1 always
}

// ---------------- WMMA f32 GEMM:  Y[n,NC] = X[n,64] @ W[64,NC] (+bias) ------
// 256 threads = 8 waves, each wave computes a 16-row tile, 128 rows per block.
// A (16x4 f32): lanes 0-15 hold K={k,k+1}, lanes 16-31 hold K={k+2,k+3}.
// B (4x16 f32): lanes 0-15 hold rows K={k,k+1}, lanes 16-31 rows K={k+2,k+3}.
// C/D (16x16 f32): VGPR v -> M=v (lanes 0-15), M=v+8 (lanes 16-31), N=lane%16.
//
// W is staged in LDS in K-pair-interleaved form so each lane's B fragment
// (two consecutive-K elements of one column) is a single aligned ds_load_b64:
//   wlds[((k>>1)*NC + n)*2 + (k&1)] = W[k*NC + n]

template <int NC, bool BIAS>
__global__ __launch_bounds__(256) void k_gemm(const float* __restrict__ X,
                                              const float* __restrict__ W,
                                              const float* __restrict__ bias,
                                              float* __restrict__ Y, int n) {
    __shared__ float wlds[64 * NC];
    for (int i = threadIdx.x; i < 64 * NC; i += 256) {
        const int r = i / NC, c = i % NC;
        wlds[(((r >> 1) * NC + c) << 1) + (r & 1)] = W[i];
    }
    __syncthreads();

    const int wave = threadIdx.x >> 5;
    const int lane = threadIdx.x & 31;
    const int l16  = lane & 15;
    const bool hi  = lane >= 16;
    const int row0 = blockIdx.x * 128 + wave * 16;
    if (row0 >= n) return;

    const float* xrow = X + (size_t)(row0 + l16) * 64;
    const v2f* wp = (const v2f*)wlds;  // pair (kpair, n) at wp[kpair*NC + n]

    v8f acc[NC / 16];
#pragma unroll
    for (int t = 0; t < NC / 16; ++t)
#pragma unroll
        for (int v = 0; v < 8; ++v) acc[t][v] = 0.0f;

#pragma unroll
    for (int k = 0; k < 64; k += 4) {
        const int ka = k + (hi ? 2 : 0);        // even -> 8B aligned
        const int kp = (ka >> 1);               // K-pair index
        v2f a = *(const v2f*)(xrow + ka);       // global_load_b64
#pragma unroll
        for (int t = 0; t < NC / 16; ++t) {
            v2f b = wp[kp * NC + t * 16 + l16]; // ds_load_b64
            acc[t] = __builtin_amdgcn_wmma_f32_16x16x4_f32(
                false, a, false, b, (short)0, acc[t], false, false);
        }
    }

#pragma unroll
    for (int t = 0; t < NC / 16; ++t) {
        const float bb = BIAS ? bias[t * 16 + l16] : 0.0f;
#pragma unroll
        for (int v = 0; v < 8; ++v) {
            const int m = v + (hi ? 8 : 0);
            Y[(size_t)(row0 + m) * NC + t * 16 + l16] = acc[t][v] + bb;
        }
    }
}

// ---------------- aggregation ----------------

// out[i,f] = h[i,f]*dinv[i]^2 + bias[f]   (self-loop term + conv bias)
__global__ void k_init_agg(const float* __restrict__ h, const float* __restrict__ dinv,
                           const float* __restrict__ bias, float* __restrict__ out, int n) {
    int i = blockIdx.x * blockDim.x + threadIdx.x;
    if (i >= n * 64) return;
    int node = i >> 6, f = i & 63;
    float di = dinv[node];
    out[i] = h[i] * di * di + bias[f];
}

// per-edge scatter: 8 threads/edge, 8 features each; fp32 global atomics
__global__ void k_edge_scatter(const int* __restrict__ src, const int* __restrict__ dst,
                               const float* __restrict__ dinv, const float* __restrict__ h,
                               float* __restrict__ out, int e) {
    long long idx = (long long)blockIdx.x * blockDim.x + threadIdx.x;
    if (idx >= (long long)e * 8) return;
    int ed = (int)(idx >> 3);
    int c  = ((int)idx & 7) << 3;
    int s = src[ed], d = dst[ed];
    float nrm = dinv[s] * dinv[d];
    const float* hs = h + (size_t)s * 64 + c;
    float* od = out + (size_t)d * 64 + c;
#pragma unroll
    for (int f = 0; f < 8; ++f) atomicAdd(od + f, hs[f] * nrm);
}

// BatchNorm eval (mean=0, var=1) + ReLU:  relu(gamma * h * rsqrt(1+eps) + beta)
__global__ void k_bnrelu(const float* __restrict__ in, const float* __restrict__ gamma,
                         const float* __restrict__ beta, float* __restrict__ out, int n) {
    int i = blockIdx.x * blockDim.x + threadIdx.x;
    if (i >= n * 64) return;
    int f = i & 63;
    float v = gamma[f] * (in[i] * 0.99999500003749969f) + beta[f];
    out[i] = v > 0.0f ? v : 0.0f;
}

// ---------------- readout ----------------

// scal layout (floats): [0..31] acc, [32] sumExp, [33] min bits (int), [34] max bits (int)
__global__ void k_init_scal(float* scal) {
    int i = threadIdx.x;
    if (i < 33) scal[i] = 0.0f;
    if (i == 33) ((int*)scal)[33] = 0x7f800000;  // +inf
    if (i == 34) ((int*)scal)[34] = 0;           // weights are positive
}

// one wave32 per node: softmax over 32 classes, entropy, raw weight + min/max
__global__ __launch_bounds__(256) void k_entropy(const float* __restrict__ logits,
                                                 float* __restrict__ wts,
                                                 int* scal_min, int* scal_max, int n) {
    int wave = threadIdx.x >> 5, lane = threadIdx.x & 31;
    int node = blockIdx.x * 8 + wave;
    if (node >= n) return;
    float v = logits[(size_t)node * 32 + lane];
    float m = v;
#pragma unroll
    for (int off = 16; off; off >>= 1) m = fmaxf(m, __shfl_xor(m, off));
    float ev = __expf(v - m);
    float s = ev;
#pragma unroll
    for (int off = 16; off; off >>= 1) s += __shfl_xor(s, off);
    float p = ev / s;
    float t = p * __logf(p + 1e-9f);
#pragma unroll
    for (int off = 16; off; off >>= 1) t += __shfl_xor(t, off);
    if (lane == 0) {
        float raw = 1.0f / (-t + 1e-10f);  // ent = -sum(p log p)
        wts[node] = raw;
        atomicMin(scal_min, __float_as_int(raw));  // positive floats: int order == fp order
        atomicMax(scal_max, __float_as_int(raw));
    }
}

// graph[o] = sum_i exp(wnorm_i) * logits[i,o]; also sum_i exp(wnorm_i)
__global__ __launch_bounds__(256) void k_weighted(const float* __restrict__ logits,
                                                  const float* __restrict__ wts,
                                                  float* scal, int n) {
    __shared__ float lacc[33];
    if (threadIdx.x < 33) lacc[threadIdx.x] = 0.0f;
    __syncthreads();
    int node = blockIdx.x * 256 + threadIdx.x;
    if (node < n) {
        float mn = __int_as_float(((const int*)scal)[33]);
        float mx = __int_as_float(((const int*)scal)[34]);
        float w = (wts[node] - mn) / (mx - mn);
        float s = __expf(w);  // wnorm in [0,1], no max-subtraction needed
        atomicAdd(&lacc[32], s);
        const float* lr = logits + (size_t)node * 32;
#pragma unroll
        for (int o = 0; o < 32; ++o) atomicAdd(&lacc[o], s * lr[o]);
    }
    __syncthreads();
    if (threadIdx.x < 33) atomicAdd(&scal[threadIdx.x], lacc[threadIdx.x]);
}

// out[h] = sum_o (acc[o]/sumExp) * lin3_w[o,h] + lin3_b[h]
__global__ void k_final(const float* __restrict__ scal, const float* __restrict__ w3,
                        const float* __restrict__ b3, float* __restrict__ out) {
    int h = threadIdx.x;
    if (h >= 64) return;
    float inv = 1.0f / scal[32];
    float r = 0.0f;
#pragma unroll
    for (int o = 0; o < 32; ++o) r += scal[o] * inv * w3[o * 64 + h];
    out[h] = r + b3[h];
}

// ---------------- launcher ----------------

extern "C" void kernel_launch(void* const* d_in, const int* in_sizes, int n_in,
                              void* d_out, int out_size, void* d_ws, size_t ws_size,
                              hipStream_t stream) {
    const int n = in_sizes[0] / 64;  // N nodes (x is [N,64])
    const int e = in_sizes[1] / 2;   // E edges (edge_index is [2,E])

    const float* x    = (const float*)d_in[0];
    const int*   ei   = (const int*)d_in[1];
    const float* W1   = (const float*)d_in[2];
    const float* b1   = (const float*)d_in[3];
    const float* gamma= (const float*)d_in[4];
    const float* beta = (const float*)d_in[5];
    const float* W2   = (const float*)d_in[6];
    const float* b2   = (const float*)d_in[7];
    const float* l2w  = (const float*)d_in[8];
    const float* l2b  = (const float*)d_in[9];
    const float* l3w  = (const float*)d_in[10];
    const float* l3b  = (const float*)d_in[11];

    const int* src = ei;
    const int* dst = ei + e;

    float* ws   = (float*)d_ws;
    float* dinv = ws;                        // N
    float* A    = dinv + n;                  // N*64
    float* B    = A + (size_t)n * 64;        // N*64
    float* L    = B + (size_t)n * 64;        // N*32
    float* wts  = L + (size_t)n * 32;        // N
    float* scal = wts + n;                   // 64

    const int nb_n   = (n + 255) / 256;
    const int nb_e   = (e + 255) / 256;
    const int nb_nf  = (int)(((size_t)n * 64 + 255) / 256);
    const int nb_e8  = (int)(((long long)e * 8 + 255) / 256);
    const int nb_gem = (n + 127) / 128;

    // degree -> dinv
    k_init_deg<<<nb_n, 256, 0, stream>>>(dinv, n);
    k_deg_scatter<<<nb_e, 256, 0, stream>>>(dst, dinv, e);
    k_dinv<<<nb_n, 256, 0, stream>>>(dinv, n);

    // layer 1: A = x@W1 ; B = agg(A) ; A = relu(bn(B))
    k_gemm<64, false><<<nb_gem, 256, 0, stream>>>(x, W1, nullptr, A, n);
    k_init_agg<<<nb_nf, 256, 0, stream>>>(A, dinv, b1, B, n);
    k_edge_scatter<<<nb_e8, 256, 0, stream>>>(src, dst, dinv, A, B, e);
    k_bnrelu<<<nb_nf, 256, 0, stream>>>(B, gamma, beta, A, n);

    // layer 2: B = A@W2 ; A = agg(B)
    k_gemm<64, false><<<nb_gem, 256, 0, stream>>>(A, W2, nullptr, B, n);
    k_init_agg<<<nb_nf, 256, 0, stream>>>(B, dinv, b2, A, n);
    k_edge_scatter<<<nb_e8, 256, 0, stream>>>(src, dst, dinv, B, A, e);

    // logits = A@lin2_w + lin2_b
    k_gemm<32, true><<<nb_gem, 256, 0, stream>>>(A, l2w, l2b, L, n);

    // readout
    k_init_scal<<<1, 64, 0, stream>>>(scal);
    k_entropy<<<(n + 7) / 8, 256, 0, stream>>>(L, wts, (int*)scal + 33, (int*)scal + 34, n);
    k_weighted<<<nb_n, 256, 0, stream>>>(L, wts, scal, n);
    k_final<<<1, 64, 0, stream>>>(scal, l3w, l3b, (float*)d_out);
}